// Network_84937273245975
// MI455X (gfx1250) — compile-verified
//
#include <hip/hip_runtime.h>

// DeLaN inverse dynamics, d=2, H1=6. Bandwidth-bound: 32 B/sample, ~84 MB
// total @ 23.3 TB/s ~= 3.6 us floor. Strategy:
//  - async-stage 12KB x-tiles into LDS (global_load_async_to_lds_b128 +
//    s_wait_asynccnt: gfx1250 async path), pure coalesced b128 streams
//  - 2 samples per thread, all algebra in v2f (lane = sample) -> lowers to
//    V_PK_FMA_F32 / V_PK_MUL_F32 (confirmed by round-2 instruction counts)
//  - weights uniform -> s_load; one b128 non-temporal store per thread
//  - single compute_pair instantiation; tail tiles stage via guarded loads
// WMMA intentionally unused: 6x2 weights + per-sample ReLU masks make
// matrix-core marshaling strictly more expensive than the packed FMAs.

typedef float v2f __attribute__((ext_vector_type(2)));
typedef float v4f __attribute__((ext_vector_type(4)));

#define DELAN_EPS 0.001f
#define BLOCK 256
#define TILE  512                 // samples per block (2 per thread)

static __device__ __forceinline__ v2f mk2(float a, float b) { v2f r; r.x = a; r.y = b; return r; }
static __device__ __forceinline__ v2f sp2(float a)          { v2f r; r.x = a; r.y = a; return r; }
static __device__ __forceinline__ v2f fma2(v2f a, v2f b, v2f c) { return __builtin_elementwise_fma(a, b, c); }

struct Wts {
    float w10[6], w11[6], bb1[6];
    float w20[6], w21[6], b20, b21;
    float w30[6], w31[6], b30, b31;
    float w4[6],  b4;
};

// Full per-sample algebra, two samples packed in the two v2f lanes.
static __device__ __forceinline__ void compute_pair(
    const Wts& W,
    v2f q0, v2f q1, v2f qd0, v2f qd1, v2f qdd0, v2f qdd1,
    v2f& tau0, v2f& tau1)
{
    v2f g0 = sp2(W.b20), g1 = sp2(W.b21);
    v2f l0 = sp2(W.b30), l1 = sp2(W.b31);
    v2f lo = sp2(W.b4);
    v2f d00 = sp2(0.f), d01 = sp2(0.f);   // sum_j W3[0,j] m_j W1[j,k]
    v2f d10 = sp2(0.f), d11 = sp2(0.f);   // sum_j W3[1,j] m_j W1[j,k]
    v2f e0  = sp2(0.f), e1  = sp2(0.f);   // sum_j W4[j]   m_j W1[j,k]
#pragma unroll
    for (int j = 0; j < 6; ++j) {
        const v2f pre = fma2(sp2(W.w10[j]), q0, fma2(sp2(W.w11[j]), q1, sp2(W.bb1[j])));
        v2f m; m.x = pre.x > 0.f ? 1.f : 0.f;
               m.y = pre.y > 0.f ? 1.f : 0.f;
        const v2f h = m * pre;                // relu(pre) via the mask (pk_mul)
        g0 = fma2(sp2(W.w20[j]), h, g0);
        g1 = fma2(sp2(W.w21[j]), h, g1);
        l0 = fma2(sp2(W.w30[j]), h, l0);
        l1 = fma2(sp2(W.w31[j]), h, l1);
        lo = fma2(sp2(W.w4[j]),  h, lo);
        const v2f p0 = m * sp2(W.w10[j]);     // masked W1 row, k=0
        const v2f p1 = m * sp2(W.w11[j]);     // masked W1 row, k=1
        d00 = fma2(sp2(W.w30[j]), p0, d00);
        d01 = fma2(sp2(W.w30[j]), p1, d01);
        d10 = fma2(sp2(W.w31[j]), p0, d10);
        d11 = fma2(sp2(W.w31[j]), p1, d11);
        e0  = fma2(sp2(W.w4[j]),  p0, e0);
        e1  = fma2(sp2(W.w4[j]),  p1, e1);
    }
    // ld = relu(l), fold ld-mask into Jacobian rows (all pk_mul)
    v2f s0, s1;
    s0.x = l0.x > 0.f ? 1.f : 0.f;  s0.y = l0.y > 0.f ? 1.f : 0.f;
    s1.x = l1.x > 0.f ? 1.f : 0.f;  s1.y = l1.y > 0.f ? 1.f : 0.f;
    const v2f a = s0 * l0;
    const v2f b = s1 * l1;
    const v2f c = lo;
    d00 = d00 * s0;  d01 = d01 * s0;
    d10 = d10 * s1;  d11 = d11 * s1;

    // time derivatives of L entries
    const v2f adot = fma2(d00, qd0, d01 * qd1);
    const v2f bdot = fma2(d10, qd0, d11 * qd1);
    const v2f cdot = fma2(e0,  qd0, e1  * qd1);

    // H = L^T L + eps I,  L = [[a,0],[c,b]]
    const v2f H00 = fma2(a, a, fma2(c, c, sp2(DELAN_EPS)));
    const v2f H01 = c * b;
    const v2f H11 = fma2(b, b, sp2(DELAN_EPS));

    // dH/dt = L dL^T + dL L^T
    const v2f two = sp2(2.0f);
    const v2f dH00 = two * (a * adot);
    const v2f dH01 = fma2(a, cdot, c * adot);
    const v2f dH11 = two * fma2(c, cdot, b * bdot);

    // quad_k = 2 a qd0^2 d0k + 2 qd0 qd1 (c d0k + a ek) + 2 qd1^2 (c ek + b d1k)
    const v2f t00 = two * (qd0 * qd0) * a;
    const v2f t01 = two * (qd0 * qd1);
    const v2f t11 = two * (qd1 * qd1);
    v2f quad0 = t00 * d00;
    quad0 = fma2(t01, fma2(c, d00, a * e0), quad0);
    quad0 = fma2(t11, fma2(c, e0, b * d10), quad0);
    v2f quad1 = t00 * d01;
    quad1 = fma2(t01, fma2(c, d01, a * e1), quad1);
    quad1 = fma2(t11, fma2(c, e1, b * d11), quad1);

    // tau = H qdd + dH/dt qd + quad + g
    tau0 = quad0 + g0;
    tau0 = fma2(dH01, qd1, fma2(dH00, qd0, tau0));
    tau0 = fma2(H01, qdd1, fma2(H00, qdd0, tau0));
    tau1 = quad1 + g1;
    tau1 = fma2(dH11, qd1, fma2(dH01, qd0, tau1));
    tau1 = fma2(H11, qdd1, fma2(H01, qdd0, tau1));
}

__global__ __launch_bounds__(BLOCK) void delan_tau_kernel(
    const float* __restrict__ x,
    const float* __restrict__ W1, const float* __restrict__ b1,
    const float* __restrict__ W2, const float* __restrict__ b2,
    const float* __restrict__ W3, const float* __restrict__ b3,
    const float* __restrict__ W4, const float* __restrict__ b4,
    float* __restrict__ out, int N)
{
    __shared__ __align__(16) float tile[TILE * 6];   // 12 KB
    const int t = threadIdx.x;
    const long base = (long)blockIdx.x * TILE;
    const bool full = (base + TILE) <= (long)N;

    // uniform weights -> SGPRs
    Wts W;
#pragma unroll
    for (int j = 0; j < 6; ++j) {
        W.w10[j] = W1[2*j];  W.w11[j] = W1[2*j + 1];  W.bb1[j] = b1[j];
        W.w20[j] = W2[j];    W.w21[j] = W2[6 + j];
        W.w30[j] = W3[j];    W.w31[j] = W3[6 + j];
        W.w4[j]  = W4[j];
    }
    W.b20 = b2[0]; W.b21 = b2[1];
    W.b30 = b3[0]; W.b31 = b3[1];
    W.b4  = b4[0];

    if (full) {
        // ---- async stage: 768 x b128 copies (3 per thread), ASYNCcnt ------
        const unsigned lds0 = (unsigned)(uintptr_t)(void*)tile;
        const char* gbase = (const char*)x + base * 24l;
#pragma unroll
        for (int i = 0; i < 3; ++i) {
            const unsigned loff = lds0 + (unsigned)(t + i * BLOCK) * 16u;
            const char* gaddr = gbase + (size_t)(t + i * BLOCK) * 16u;
            asm volatile("global_load_async_to_lds_b128 %0, %1, off"
                         :: "v"(loff), "v"(gaddr) : "memory");
        }
        // stream-prefetch the next block's tile (gfx1250 global_prefetch_b8)
        __builtin_prefetch(gbase + (size_t)TILE * 24 + (size_t)t * 48, 0, 0);
        asm volatile("s_wait_asynccnt 0x0" ::: "memory");
    } else {
        // tail tile (not taken for N % TILE == 0): guarded staging
        const long limit = (long)N * 6;            // floats in x
#pragma unroll
        for (int i = 0; i < 3; ++i) {
            const long f0 = base * 6 + (long)(t + i * BLOCK) * 4;  // float idx
            v4f v = {0.f, 0.f, 0.f, 0.f};
            if (f0 + 4 <= limit) {
                v = *(const v4f*)(x + f0);
            } else {
#pragma unroll
                for (int k = 0; k < 4; ++k)
                    if (f0 + k < limit) v[k] = x[f0 + k];
            }
            *(v4f*)&tile[(t + i * BLOCK) * 4] = v;
        }
    }
    __syncthreads();

    // samples (base+2t, base+2t+1): 48 contiguous bytes in LDS
    const float* tp = &tile[t * 12];
    const v2f q0   = mk2(tp[0], tp[6]);
    const v2f q1   = mk2(tp[1], tp[7]);
    const v2f qd0  = mk2(tp[2], tp[8]);
    const v2f qd1  = mk2(tp[3], tp[9]);
    const v2f qdd0 = mk2(tp[4], tp[10]);
    const v2f qdd1 = mk2(tp[5], tp[11]);

    v2f tau0, tau1;
    compute_pair(W, q0, q1, qd0, qd1, qdd0, qdd1, tau0, tau1);

    if (full) {
        // both samples' tau are contiguous: one NT b128 store
        v4f o; o.x = tau0.x; o.y = tau1.x; o.z = tau0.y; o.w = tau1.y;
        __builtin_nontemporal_store(o, (v4f*)out + (base >> 1) + t);
    } else {
        const long nA = base + 2l * t;
        if (nA < N) {
            out[2*nA]     = tau0.x;
            out[2*nA + 1] = tau1.x;
            if (nA + 1 < N) {
                out[2*nA + 2] = tau0.y;
                out[2*nA + 3] = tau1.y;
            }
        }
    }
}

extern "C" void kernel_launch(void* const* d_in, const int* in_sizes, int n_in,
                              void* d_out, int out_size, void* d_ws, size_t ws_size,
                              hipStream_t stream) {
    const float* x  = (const float*)d_in[0];
    const float* W1 = (const float*)d_in[1];
    const float* b1 = (const float*)d_in[2];
    const float* W2 = (const float*)d_in[3];
    const float* b2 = (const float*)d_in[4];
    const float* W3 = (const float*)d_in[5];
    const float* b3 = (const float*)d_in[6];
    const float* W4 = (const float*)d_in[7];
    const float* b4 = (const float*)d_in[8];

    const int N = in_sizes[0] / 6;                 // x is (N, 6)
    const int grid = (N + TILE - 1) / TILE;

    delan_tau_kernel<<<grid, BLOCK, 0, stream>>>(
        x, W1, b1, W2, b2, W3, b3, W4, b4, (float*)d_out, N);
}